// PoleTransportDetector_33011118637130
// MI455X (gfx1250) — compile-verified
//
#include <hip/hip_runtime.h>
#include <cmath>

// Problem dimensions (match reference)
#define NB   32          // batch
#define NA   16          // rx antennas
#define NL   14          // OFDM symbols
#define NK   600         // subcarriers
#define NM   32          // reservoir poles per direction
#define NR   4           // low-rank mixing rank
#define NN   (NL*NK)     // 8400 grid points per (b, antenna)
#define NPIL 1200        // pilot tones (2 symbols * 600)
#define NF   145         // feature count
#define NFP  160         // padded feature count (multiple of 16)
#define PILA 2
#define PILB 11
#define RIDGE 1e-3f
#define WEPS  1e-3f

typedef float v8f __attribute__((ext_vector_type(8)));
typedef float v2f __attribute__((ext_vector_type(2)));

// ---------------- complex helpers (float2 = {re, im}) ----------------
__device__ __forceinline__ float2 cmul(float2 a, float2 b) {
    return make_float2(a.x*b.x - a.y*b.y, a.x*b.y + a.y*b.x);
}
__device__ __forceinline__ float2 cmadd(float2 a, float2 b, float2 c) {
    // a*b + c
    return make_float2(fmaf(a.x, b.x, fmaf(-a.y, b.y, c.x)),
                       fmaf(a.x, b.y, fmaf( a.y, b.x, c.y)));
}
__device__ __forceinline__ float2 cadd(float2 a, float2 b) { return make_float2(a.x+b.x, a.y+b.y); }
__device__ __forceinline__ float2 csub(float2 a, float2 b) { return make_float2(a.x-b.x, a.y-b.y); }

// ---------------- fp32 WMMA 16x16x4 wrapper ----------------
// A fragment: lane m (0-15) holds A[m][0],A[m][1] in (a0,a1); lane m+16 holds A[m][2],A[m][3].
// B fragment: lane n (0-15) holds B[0][n],B[1][n]; lane n+16 holds B[2][n],B[3][n].
// C/D: VGPR v holds (M = v + 8*(lane>=16), N = lane&15).
__device__ __forceinline__ v8f wmma4(float a0, float a1, float b0, float b1, v8f c) {
#if __has_builtin(__builtin_amdgcn_wmma_f32_16x16x4_f32)
    v2f av = {a0, a1};
    v2f bv = {b0, b1};
    return __builtin_amdgcn_wmma_f32_16x16x4_f32(false, av, false, bv, (short)0, c, false, false);
#else
    // Correct lane-shuffle fallback with identical operand layout.
    int lane = (int)(threadIdx.x & 31u);
    int n  = lane & 15;
    int hi = lane >> 4;
#pragma unroll
    for (int v = 0; v < 8; ++v) {
        int m = v + hi * 8;
        float s = __shfl(a0, m)      * __shfl(b0, n)
                + __shfl(a1, m)      * __shfl(b1, n)
                + __shfl(a0, m + 16) * __shfl(b0, n + 16)
                + __shfl(a1, m + 16) * __shfl(b1, n + 16);
        c[v] += s;
    }
    return c;
#endif
}

// ---------------- async global->LDS copy (CDNA5 ASYNCcnt path) ----------------
// Probe-derived signature: (AS1 v4i* gsrc, AS3 v4i* lds, int offset, int cpol)
#if defined(__HIP_DEVICE_COMPILE__) && __has_builtin(__builtin_amdgcn_global_load_async_to_lds_b128)
#define HAVE_ASYNC_LDS 1
typedef int vec4i __attribute__((vector_size(16)));
__device__ __forceinline__ void async_copy_b128(const void* g, void* l) {
    __builtin_amdgcn_global_load_async_to_lds_b128(
        (__attribute__((address_space(1))) vec4i*)(g),
        (__attribute__((address_space(3))) vec4i*)(l),
        0, 0);
}
__device__ __forceinline__ void async_wait0() {
#if __has_builtin(__builtin_amdgcn_s_wait_asynccnt)
    __builtin_amdgcn_s_wait_asynccnt(0);
#else
    asm volatile("s_wait_asynccnt 0x0" ::: "memory");
#endif
}
#endif

// =====================================================================
// 1) Pilot spatial covariance:  Rcov[b][a][c] = sum_n yp[a,n] conj(yp[c,n]) / NPIL
//    + WHITE_EPS * (trace/NA) * I
// grid: NB blocks, 256 threads (one thread per (a,c))
// =====================================================================
__global__ void cov_kernel(const float2* __restrict__ y, float2* __restrict__ Rcov) {
    int b = blockIdx.x;
    int a = threadIdx.x >> 4;
    int c = threadIdx.x & 15;
    const float2* ya = y + ((size_t)b*NA + a)*(size_t)NN;
    const float2* yc = y + ((size_t)b*NA + c)*(size_t)NN;
    float2 acc = make_float2(0.f, 0.f);
#pragma unroll
    for (int lp = 0; lp < 2; ++lp) {
        int l = lp ? PILB : PILA;
        const float2* ra = ya + (size_t)l*NK;
        const float2* rc = yc + (size_t)l*NK;
        for (int k = 0; k < NK; ++k) {
            float2 u = ra[k], v = rc[k];
            acc.x += u.x*v.x + u.y*v.y;   // u * conj(v)
            acc.y += u.y*v.x - u.x*v.y;
        }
    }
    acc.x *= (1.0f / NPIL);
    acc.y *= (1.0f / NPIL);
    __shared__ float diag[NA];
    __shared__ float eps;
    if (a == c) diag[a] = acc.x;
    __syncthreads();
    if (threadIdx.x == 0) {
        float s = 0.f;
        for (int i = 0; i < NA; ++i) s += diag[i];
        eps = (s / NA) * WEPS;
    }
    __syncthreads();
    if (a == c) acc.x += eps;
    Rcov[((size_t)b*NA + a)*NA + c] = acc;
}

// =====================================================================
// 2) Per-batch complex Cholesky + explicit lower-triangular inverse
// grid: 1 block of 32 threads (one thread per batch)
// =====================================================================
__global__ void chol_inv_kernel(const float2* __restrict__ Rcov, float2* __restrict__ Linv) {
    int b = blockIdx.x * blockDim.x + threadIdx.x;
    if (b >= NB) return;
    const float2* Rb = Rcov + (size_t)b*NA*NA;
    float2 Lm[NA][NA];
    float  dinv[NA];
    for (int j = 0; j < NA; ++j) {
        float s = Rb[j*NA + j].x;
        for (int k = 0; k < j; ++k) { float2 v = Lm[j][k]; s -= v.x*v.x + v.y*v.y; }
        float d = sqrtf(fmaxf(s, 1e-20f));
        Lm[j][j] = make_float2(d, 0.f);
        dinv[j]  = 1.f / d;
        for (int i = j + 1; i < NA; ++i) {
            float2 s2 = Rb[i*NA + j];
            for (int k = 0; k < j; ++k) {
                float2 aa = Lm[i][k], cc = Lm[j][k];
                s2.x -= aa.x*cc.x + aa.y*cc.y;     // a * conj(c)
                s2.y -= aa.y*cc.x - aa.x*cc.y;
            }
            Lm[i][j] = make_float2(s2.x*dinv[j], s2.y*dinv[j]);
        }
    }
    float2 Inv[NA][NA];
    for (int j = 0; j < NA; ++j) {
        for (int i = 0; i < j; ++i) Inv[i][j] = make_float2(0.f, 0.f);
        Inv[j][j] = make_float2(dinv[j], 0.f);
        for (int i = j + 1; i < NA; ++i) {
            float2 s = make_float2(0.f, 0.f);
            for (int k = j; k < i; ++k) s = cmadd(Lm[i][k], Inv[k][j], s);
            Inv[i][j] = make_float2(-s.x*dinv[i], -s.y*dinv[i]);
        }
    }
    float2* O = Linv + (size_t)b*NA*NA;
    for (int i = 0; i < NA; ++i)
        for (int j = 0; j < NA; ++j)
            O[i*NA + j] = Inv[i][j];
}

// =====================================================================
// 3) Whitening via WMMA: yw[b] = Linv[b] (16x16 complex) @ y[b] (16 x 8400)
// grid: (NN/16, NB), block 32 (one wave = one 16x16 output tile)
// =====================================================================
__global__ void whiten_kernel(const float2* __restrict__ y, const float2* __restrict__ Linv,
                              float2* __restrict__ yw) {
    int b  = blockIdx.y;
    int n0 = blockIdx.x * 16;
    const float2* Am = Linv + (size_t)b*NA*NA;
    const float2* Yb = y    + (size_t)b*NA*NN;
    int lane = (int)threadIdx.x;
    int nn = lane & 15;
    int lo = (lane >> 4) * 2;
    int hi = lane >> 4;
    v8f cre = {0,0,0,0,0,0,0,0};
    v8f cim = {0,0,0,0,0,0,0,0};
#pragma unroll
    for (int c = 0; c < 4; ++c) {
        int k0 = 4*c;
        float2 a0 = Am[nn*NA + k0 + lo];
        float2 a1 = Am[nn*NA + k0 + lo + 1];
        float2 b0 = Yb[(size_t)(k0 + lo    )*NN + n0 + nn];
        float2 b1 = Yb[(size_t)(k0 + lo + 1)*NN + n0 + nn];
        cre = wmma4( a0.x,  a1.x, b0.x, b1.x, cre);   // Ar*Yr
        cre = wmma4(-a0.y, -a1.y, b0.y, b1.y, cre);   // -Ai*Yi
        cim = wmma4( a0.x,  a1.x, b0.y, b1.y, cim);   // Ar*Yi
        cim = wmma4( a0.y,  a1.y, b0.x, b1.x, cim);   // Ai*Yr
    }
    float2* O = yw + (size_t)b*NA*NN;
#pragma unroll
    for (int v = 0; v < 8; ++v) {
        int m = v + hi*8;
        O[(size_t)m*NN + n0 + nn] = make_float2(cre[v], cim[v]);
    }
}

// =====================================================================
// 4) Projections via WMMA: u[b][n][m] = sum_a Proj[m][a] * yw[b][a][n]
// grid: (NN/16, 4, NB); blockIdx.y: bit0 = row tile (m0), bit1 = direction
// =====================================================================
__global__ void proj_kernel(const float2* __restrict__ Bf, const float2* __restrict__ Bb,
                            const float2* __restrict__ yw,
                            float2* __restrict__ uf, float2* __restrict__ ub) {
    int b   = blockIdx.z;
    int dir = blockIdx.y >> 1;
    int m0  = (blockIdx.y & 1) * 16;
    int n0  = blockIdx.x * 16;
    const float2* Pm = dir ? Bb : Bf;            // [32][16]
    const float2* Yb = yw + (size_t)b*NA*NN;
    int lane = (int)threadIdx.x;
    int nn = lane & 15;
    int lo = (lane >> 4) * 2;
    int hi = lane >> 4;
    v8f cre = {0,0,0,0,0,0,0,0};
    v8f cim = {0,0,0,0,0,0,0,0};
#pragma unroll
    for (int c = 0; c < 4; ++c) {
        int k0 = 4*c;
        float2 a0 = Pm[(m0 + nn)*NA + k0 + lo];
        float2 a1 = Pm[(m0 + nn)*NA + k0 + lo + 1];
        float2 b0 = Yb[(size_t)(k0 + lo    )*NN + n0 + nn];
        float2 b1 = Yb[(size_t)(k0 + lo + 1)*NN + n0 + nn];
        cre = wmma4( a0.x,  a1.x, b0.x, b1.x, cre);
        cre = wmma4(-a0.y, -a1.y, b0.y, b1.y, cre);
        cim = wmma4( a0.x,  a1.x, b0.y, b1.y, cim);
        cim = wmma4( a0.y,  a1.y, b0.x, b1.x, cim);
    }
    float2* U = (dir ? ub : uf) + (size_t)b*NN*NM;
#pragma unroll
    for (int v = 0; v < 8; ++v) {
        int m = m0 + v + hi*8;
        U[(size_t)(n0 + nn)*NM + m] = make_float2(cre[v], cim[v]);
    }
}

// =====================================================================
// 5) Bidirectional pole recursion across K (in-place u -> s)
// grid: NB*NL blocks, 64 threads (lanes 0-31 forward, 32-63 backward)
// Latency-bound dependent chain: prefetch 16 steps ahead to keep HBM busy.
// =====================================================================
__global__ void scan_kernel(const float2* __restrict__ poles,
                            float2* __restrict__ uf, float2* __restrict__ ub) {
    int bl = blockIdx.x;
    int b = bl / NL, l = bl % NL;
    int t = (int)threadIdx.x;
    int m = t & 31;
    int bwd = t >> 5;
    float2 p = poles[m];
    float2* U = (bwd ? ub : uf) + ((size_t)b*NN + (size_t)l*NK)*NM + m;
    float2 s = make_float2(0.f, 0.f);
    if (!bwd) {
        for (int k = 0; k < NK; ++k) {
            if ((k & 15) == 0 && k + 16 < NK)
                __builtin_prefetch(U + (size_t)(k + 16)*NM, 0, 0);
            float2 u = U[(size_t)k*NM];
            s = cmadd(p, s, u);
            U[(size_t)k*NM] = s;
        }
    } else {
        for (int k = NK - 1; k >= 0; --k) {
            if ((k & 15) == 15 && k - 16 >= 0)
                __builtin_prefetch(U + (size_t)(k - 16)*NM, 0, 0);
            float2 u = U[(size_t)k*NM];
            s = cmadd(p, s, u);
            U[(size_t)k*NM] = s;
        }
    }
}

// =====================================================================
// 6) Build pilot feature matrix Phi_p [B][1200][160] (zero-padded to 160)
// grid: (NPIL, NB), block NFP=160
// Feature order: sf(32) sb(32) mixed(32) prev(16) cur(16) next(16) bias(1) pad(15)
// =====================================================================
__global__ void build_phi_kernel(const float2* __restrict__ sf, const float2* __restrict__ sb,
                                 const float2* __restrict__ yw,
                                 const float2* __restrict__ Uc, const float2* __restrict__ Vc,
                                 float2* __restrict__ Phi) {
    int n = blockIdx.x, b = blockIdx.y;
    int f = (int)threadIdx.x;
    int l = (n < NK) ? PILA : PILB;
    int k = n % NK;
    size_t nf = (size_t)l*NK + k;
    const float2* SF = sf + ((size_t)b*NN + nf)*NM;
    const float2* SB = sb + ((size_t)b*NN + nf)*NM;
    __shared__ float2 ss[NM];
    __shared__ float2 tt[NR];
    if (f < NM) ss[f] = cadd(SF[f], SB[f]);
    __syncthreads();
    if (f < NR) {
        float2 acc = make_float2(0.f, 0.f);
        for (int m = 0; m < NM; ++m) acc = cmadd(Vc[f*NM + m], ss[m], acc);
        tt[f] = acc;
    }
    __syncthreads();
    float2 val = make_float2(0.f, 0.f);
    if (f < 32)       val = SF[f];
    else if (f < 64)  val = SB[f - 32];
    else if (f < 96) {
        int m = f - 64;
        float2 acc = make_float2(0.f, 0.f);
        for (int r = 0; r < NR; ++r) acc = cmadd(Uc[r*NM + m], tt[r], acc);
        val = acc;
    }
    else if (f < 112) { int a = f - 96;  if (k > 0)      val = yw[((size_t)b*NA + a)*NN + nf - 1]; }
    else if (f < 128) { int a = f - 112;                 val = yw[((size_t)b*NA + a)*NN + nf];     }
    else if (f < 144) { int a = f - 128; if (k < NK - 1) val = yw[((size_t)b*NA + a)*NN + nf + 1]; }
    else if (f == 144) val = make_float2(1.f, 0.f);
    Phi[((size_t)b*NPIL + n)*NFP + f] = val;
}

// =====================================================================
// 7) Gram via WMMA, LDS-staged: G[b] = Phi_p^H Phi_p + RIDGE*I
//    G_re = Pr^T Pr + Pi^T Pi ;  G_im = Pr^T Pi - Pi^T Pr
// Block = 128 threads (4 waves) computing a 32x32 output region (2x2 tiles).
// Each K-chunk of 16 pilot rows is staged once into LDS, then all 4 waves
// feed their WMMAs from LDS. Staging uses the CDNA5 async global->LDS path
// (ASYNCcnt) when the builtin is available, else plain b128 copies.
// Row stride 40 float2 (80 banks) makes the half-wave fragment reads
// (rows k, k+2) land 32 banks apart: conflict-free on 64-bank LDS.
// grid: (NFP/32, NFP/32, NB), block 128
// =====================================================================
#define GKC  16    // staged pilot rows per chunk
#define GROW 40    // LDS row stride in float2 (32 data + 8 pad)

__global__ __launch_bounds__(128) void gram_kernel(const float2* __restrict__ Phi,
                                                   float2* __restrict__ G) {
    int b  = blockIdx.z;
    int f0 = blockIdx.x * 32;
    int g0 = blockIdx.y * 32;
    const float2* P = Phi + (size_t)b*NPIL*NFP;

    __shared__ float2 Pa[GKC * GROW];   // rows n, cols f0..f0+31
    __shared__ float2 Pb[GKC * GROW];   // rows n, cols g0..g0+31

    int tid  = (int)threadIdx.x;
    int wave = tid >> 5;
    int lane = tid & 31;
    int nn = lane & 15;
    int lo = (lane >> 4) * 2;
    int hi = lane >> 4;
    int fi = (wave >> 1) * 16;    // wave's f sub-tile offset within the 32-block
    int gi = (wave & 1) * 16;     // wave's g sub-tile offset

    // staging pattern: thread -> (row 0..15, 4 consecutive float2 = 2 float4)
    int sr = tid >> 3;
    int sc = (tid & 7) * 4;

    v8f cre = {0,0,0,0,0,0,0,0};
    v8f cim = {0,0,0,0,0,0,0,0};

    for (int c0 = 0; c0 < NPIL; c0 += GKC) {
        __syncthreads();
        const float4* gA = (const float4*)(P + (size_t)(c0 + sr)*NFP + f0 + sc);
        const float4* gB = (const float4*)(P + (size_t)(c0 + sr)*NFP + g0 + sc);
        float4* lA = (float4*)(Pa + sr*GROW + sc);
        float4* lB = (float4*)(Pb + sr*GROW + sc);
#ifdef HAVE_ASYNC_LDS
        async_copy_b128(gA + 0, lA + 0);
        async_copy_b128(gA + 1, lA + 1);
        async_copy_b128(gB + 0, lB + 0);
        async_copy_b128(gB + 1, lB + 1);
#else
        lA[0] = gA[0]; lA[1] = gA[1];
        lB[0] = gB[0]; lB[1] = gB[1];
#endif
        if (c0 + GKC < NPIL) {   // speculative prefetch of next chunk
            __builtin_prefetch(P + (size_t)(c0 + GKC + sr)*NFP + f0 + sc, 0, 0);
            __builtin_prefetch(P + (size_t)(c0 + GKC + sr)*NFP + g0 + sc, 0, 0);
        }
#ifdef HAVE_ASYNC_LDS
        async_wait0();           // own wave's ASYNCcnt == 0 (LDS writes landed)
#endif
        __syncthreads();
#pragma unroll
        for (int kk = 0; kk < GKC; kk += 4) {
            float2 a0 = Pa[(kk + lo    )*GROW + fi + nn];
            float2 a1 = Pa[(kk + lo + 1)*GROW + fi + nn];
            float2 b0 = Pb[(kk + lo    )*GROW + gi + nn];
            float2 b1 = Pb[(kk + lo + 1)*GROW + gi + nn];
            cre = wmma4( a0.x,  a1.x, b0.x, b1.x, cre);   // Pr^T Pr
            cre = wmma4( a0.y,  a1.y, b0.y, b1.y, cre);   // Pi^T Pi
            cim = wmma4( a0.x,  a1.x, b0.y, b1.y, cim);   // Pr^T Pi
            cim = wmma4(-a0.y, -a1.y, b0.x, b1.x, cim);   // -Pi^T Pr
        }
    }
    float2* Gb = G + (size_t)b*NFP*NFP;
#pragma unroll
    for (int v = 0; v < 8; ++v) {
        int m = v + hi*8;
        int fr = f0 + fi + m, gc = g0 + gi + nn;
        float re = cre[v] + ((fr == gc) ? RIDGE : 0.f);
        Gb[(size_t)fr*NFP + gc] = make_float2(re, cim[v]);
    }
}

// =====================================================================
// 8) rhs[b][f] = sum_n conj(Phi_p[b,n,f]) * xp[n]
// grid: NB, block NFP
// =====================================================================
__global__ void rhs_kernel(const float2* __restrict__ Phi, const float2* __restrict__ xp,
                           float2* __restrict__ rhs) {
    int b = blockIdx.x;
    int f = (int)threadIdx.x;
    const float2* P = Phi + (size_t)b*NPIL*NFP + f;
    float2 acc = make_float2(0.f, 0.f);
    for (int n = 0; n < NPIL; ++n) {
        float2 ph = P[(size_t)n*NFP];
        float2 x  = xp[n];
        acc.x += ph.x*x.x + ph.y*x.y;   // conj(ph) * x
        acc.y += ph.x*x.y - ph.y*x.x;
    }
    rhs[(size_t)b*NFP + f] = acc;
}

// =====================================================================
// 9) Per-batch solve (G + ridge) W = rhs via Gaussian elimination
//    (Hermitian PD + ridge => no pivoting needed; padded pivots = RIDGE)
// grid: NB, block 256
// =====================================================================
__global__ void solve_kernel(float2* __restrict__ G, const float2* __restrict__ rhs,
                             float2* __restrict__ W) {
    int b = blockIdx.x;
    int tid = (int)threadIdx.x;
    float2* Gb = G + (size_t)b*NFP*NFP;
    float2* w  = W + (size_t)b*NFP;
    const float2* r = rhs + (size_t)b*NFP;
    for (int i = tid; i < NFP; i += blockDim.x) w[i] = r[i];
    __syncthreads();
    __shared__ float2 pinv_s;
    for (int i = 0; i < NFP; ++i) {
        if (tid == 0) {
            float2 pv = Gb[(size_t)i*NFP + i];
            float d = pv.x*pv.x + pv.y*pv.y;
            d = (d > 1e-30f) ? d : 1e-30f;
            pinv_s = make_float2(pv.x/d, -pv.y/d);
            w[i] = cmul(w[i], pinv_s);
        }
        __syncthreads();
        float2 pinv = pinv_s;
        for (int c = i + 1 + tid; c < NFP; c += blockDim.x)
            Gb[(size_t)i*NFP + c] = cmul(Gb[(size_t)i*NFP + c], pinv);
        __syncthreads();
        float2 wi = w[i];
        for (int rr = i + 1 + tid; rr < NFP; rr += blockDim.x) {
            float2 fct = Gb[(size_t)rr*NFP + i];
            for (int c = i + 1; c < NFP; ++c)
                Gb[(size_t)rr*NFP + c] = csub(Gb[(size_t)rr*NFP + c],
                                              cmul(fct, Gb[(size_t)i*NFP + c]));
            w[rr] = csub(w[rr], cmul(fct, wi));
        }
        __syncthreads();
    }
    if (tid == 0) {   // back substitution (unit diagonal upper triangular)
        for (int i = NFP - 1; i >= 0; --i) {
            float2 acc = w[i];
            for (int c = i + 1; c < NFP; ++c) acc = csub(acc, cmul(Gb[(size_t)i*NFP + c], w[c]));
            w[i] = acc;
        }
    }
}

// =====================================================================
// 10) Fold rank-4 mixing into effective reservoir weights:
//     wU[r] = sum_m Uc[r][m] W[64+m]; wmix[m] = sum_r Vc[r][m] wU[r]
//     Wsf[m] = W[m] + wmix[m];  Wsb[m] = W[32+m] + wmix[m]
// grid: NB, block NM
// =====================================================================
__global__ void prep_weights_kernel(const float2* __restrict__ W,
                                    const float2* __restrict__ Uc, const float2* __restrict__ Vc,
                                    float2* __restrict__ Wsf, float2* __restrict__ Wsb) {
    int b = blockIdx.x;
    int m = (int)threadIdx.x;
    __shared__ float2 wU[NR];
    const float2* Wb = W + (size_t)b*NFP;
    if (m < NR) {
        float2 acc = make_float2(0.f, 0.f);
        for (int j = 0; j < NM; ++j) acc = cmadd(Uc[m*NM + j], Wb[64 + j], acc);
        wU[m] = acc;
    }
    __syncthreads();
    float2 wmix = make_float2(0.f, 0.f);
    for (int r = 0; r < NR; ++r) wmix = cmadd(Vc[r*NM + m], wU[r], wmix);
    Wsf[(size_t)b*NM + m] = cadd(Wb[m],      wmix);
    Wsb[(size_t)b*NM + m] = cadd(Wb[32 + m], wmix);
}

// =====================================================================
// 11) Readout on full grid: xhat = Phi . W (features recomputed on the fly)
// one thread per (b,l,k)
// =====================================================================
__global__ void readout_kernel(const float2* __restrict__ sf, const float2* __restrict__ sb,
                               const float2* __restrict__ yw, const float2* __restrict__ W,
                               const float2* __restrict__ Wsf, const float2* __restrict__ Wsb,
                               float* __restrict__ out) {
    int idx = blockIdx.x * blockDim.x + (int)threadIdx.x;
    if (idx >= NB*NN) return;
    int b = idx / NN;
    int n = idx % NN;
    int k = n % NK;
    const float2* SF = sf + ((size_t)b*NN + n)*NM;
    const float2* SB = sb + ((size_t)b*NN + n)*NM;
    const float2* wf = Wsf + (size_t)b*NM;
    const float2* wb = Wsb + (size_t)b*NM;
    const float2* Wb = W + (size_t)b*NFP;
    float2 acc = Wb[144];                       // bias feature (1 + 0i)
    for (int m = 0; m < NM; ++m) {
        acc = cmadd(SF[m], wf[m], acc);
        acc = cmadd(SB[m], wb[m], acc);
    }
    const float2* Yb = yw + (size_t)b*NA*NN;
    for (int a = 0; a < NA; ++a) {
        const float2* ya = Yb + (size_t)a*NN + n;
        acc = cmadd(ya[0], Wb[112 + a], acc);                  // cur
        if (k > 0)      acc = cmadd(ya[-1], Wb[96  + a], acc); // prev
        if (k < NK - 1) acc = cmadd(ya[ 1], Wb[128 + a], acc); // next
    }
    out[2*idx]     = acc.x;
    out[2*idx + 1] = acc.y;
}

// =====================================================================
extern "C" void kernel_launch(void* const* d_in, const int* in_sizes, int n_in,
                              void* d_out, int out_size, void* d_ws, size_t ws_size,
                              hipStream_t stream) {
    (void)in_sizes; (void)n_in; (void)out_size; (void)ws_size;
    const float2* y  = (const float2*)d_in[0];   // [B,A,L,K] complex
    const float2* pf = (const float2*)d_in[1];   // [M,A]
    const float2* pb = (const float2*)d_in[2];   // [M,A]
    const float2* po = (const float2*)d_in[3];   // [M]
    const float2* mu = (const float2*)d_in[4];   // [R,M]
    const float2* mv = (const float2*)d_in[5];   // [R,M]
    const float2* xp = (const float2*)d_in[6];   // [NPIL]
    float* out = (float*)d_out;

    char* ws = (char*)d_ws;
    size_t off = 0;
    auto take = [&](size_t elems) -> float2* {
        float2* p = (float2*)(ws + off);
        off += ((elems * sizeof(float2)) + 255) & ~(size_t)255;
        return p;
    };
    float2* Rcov = take((size_t)NB*NA*NA);
    float2* Linv = take((size_t)NB*NA*NA);
    float2* yw   = take((size_t)NB*NA*NN);
    float2* uf   = take((size_t)NB*NN*NM);    // becomes sf in-place after scan
    float2* ub   = take((size_t)NB*NN*NM);    // becomes sb in-place after scan
    float2* Phi  = take((size_t)NB*NPIL*NFP);
    float2* G    = take((size_t)NB*NFP*NFP);
    float2* rhs  = take((size_t)NB*NFP);
    float2* W    = take((size_t)NB*NFP);
    float2* Wsf  = take((size_t)NB*NM);
    float2* Wsb  = take((size_t)NB*NM);

    cov_kernel        <<<NB, 256, 0, stream>>>(y, Rcov);
    chol_inv_kernel   <<<1, 32, 0, stream>>>(Rcov, Linv);
    whiten_kernel     <<<dim3(NN/16, NB), 32, 0, stream>>>(y, Linv, yw);
    proj_kernel       <<<dim3(NN/16, 4, NB), 32, 0, stream>>>(pf, pb, yw, uf, ub);
    scan_kernel       <<<NB*NL, 64, 0, stream>>>(po, uf, ub);
    build_phi_kernel  <<<dim3(NPIL, NB), NFP, 0, stream>>>(uf, ub, yw, mu, mv, Phi);
    gram_kernel       <<<dim3(NFP/32, NFP/32, NB), 128, 0, stream>>>(Phi, G);
    rhs_kernel        <<<NB, NFP, 0, stream>>>(Phi, xp, rhs);
    solve_kernel      <<<NB, 256, 0, stream>>>(G, rhs, W);
    prep_weights_kernel<<<NB, NM, 0, stream>>>(W, mu, mv, Wsf, Wsb);
    readout_kernel    <<<(NB*NN + 255)/256, 256, 0, stream>>>(uf, ub, yw, W, Wsf, Wsb, out);
}